// Encoder_27917287424660
// MI455X (gfx1250) — compile-verified
//
#include <hip/hip_runtime.h>
#include <hip/hip_bf16.h>

// ---------------------------------------------------------------------------
// CDNA5 (gfx1250) wave32 WMMA helpers: V_WMMA_F32_16X16X4_F32 (exact fp32)
// ---------------------------------------------------------------------------
typedef float v2f __attribute__((ext_vector_type(2)));
typedef float v8f __attribute__((ext_vector_type(8)));

// A (16x4 f32): lane = h*16 + m ; v[j] = A[m][2*h + j]
__device__ __forceinline__ v2f ldA(const float* A, int lda, int r0, int k0, int lane) {
  int m = lane & 15, h = lane >> 4;
  const float* p = A + (size_t)(r0 + m) * lda + (k0 + 2 * h);
  v2f v; v.x = p[0]; v.y = p[1];
  return v;
}

// B (4x16 f32, K-major rows): lane = h*16 + n ; v[j] = B[2*h + j][n]
__device__ __forceinline__ v2f ldB(const float* Bp, int ldb, int k0, int n0, int lane) {
  int n = lane & 15, h = lane >> 4;
  const float* p = Bp + (size_t)(k0 + 2 * h) * ldb + (n0 + n);
  v2f v; v.x = p[0]; v.y = p[(size_t)ldb];
  return v;
}

// C/D (16x16 f32): VGPR r -> M = r + 8*h, N = lane&15
__device__ __forceinline__ void stC(float* D, int ldd, int r0, int n0, int lane, v8f acc,
                                    const float* bias, float alpha, bool act) {
  int n = lane & 15, h = lane >> 4;
  float bv = bias ? bias[n0 + n] : 0.0f;
#pragma unroll
  for (int r = 0; r < 8; ++r) {
    int m = r + 8 * h;
    float v = acc[r] + bv;
    if (act) v = (v >= 0.0f) ? v : alpha * v;
    D[(size_t)(r0 + m) * ldd + (n0 + n)] = v;
  }
}

// NR row-tiles sharing one B fragment per K-step (B loaded once, NR WMMAs).
// Keeps NR independent accumulator chains in flight to cover WMMA latency.
template <int NR>
__device__ __forceinline__ void gemm_rows(const float* A, int lda, const float* B,
                                          int ldb, int r0, int n0, int K, int lane,
                                          v8f* acc) {
  for (int k = 0; k < K; k += 4) {
    v2f bfr = ldB(B, ldb, k, n0, lane);
#pragma unroll
    for (int r = 0; r < NR; ++r) {
      v2f afr = ldA(A, lda, r0 + r * 16, k, lane);
      acc[r] = __builtin_amdgcn_wmma_f32_16x16x4_f32(false, afr, false, bfr, (short)0,
                                                     acc[r], false, false);
    }
  }
}

// ---------------------------------------------------------------------------
// Problem constants
// ---------------------------------------------------------------------------
#define BB   16
#define NN   512
#define CIN  512
#define HID  128
#define BCH  512
#define LPI  640
#define KM   16

// ---------------------------------------------------------------------------
// Kernel 1: fused 2-layer MLP (512 -> 128 lrelu -> 128) for wc / us.
// One block = 128 rows, 8 waves; each wave owns one 16-wide column strip and
// runs all 8 row tiles against a shared B fragment (8x B reuse).
// ---------------------------------------------------------------------------
__global__ __launch_bounds__(256) void feat_mlp(const float* __restrict__ X,
                                                const float* __restrict__ W1,
                                                const float* __restrict__ B1,
                                                const float* __restrict__ Al,
                                                const float* __restrict__ W2,
                                                const float* __restrict__ B2,
                                                float* __restrict__ OUT) {
  extern __shared__ float sm[];
  float* H = sm;                       // 128 x 132
  const int LH = 132;
  const int lane = threadIdx.x & 31, w = threadIdx.x >> 5;
  const int row0 = blockIdx.x * 128;
  const float alpha = Al[0];
  const v8f z = {0.f, 0.f, 0.f, 0.f, 0.f, 0.f, 0.f, 0.f};

  // Layer 1: X(128x512) @ W1(512x128) -> H (lrelu + bias). Wave w: ct = w.
  {
    v8f acc[8];
#pragma unroll
    for (int r = 0; r < 8; ++r) acc[r] = z;
    gemm_rows<8>(X + (size_t)row0 * CIN, CIN, W1, HID, 0, w * 16, CIN, lane, acc);
#pragma unroll
    for (int r = 0; r < 8; ++r)
      stC(H, LH, r * 16, w * 16, lane, acc[r], B1, alpha, true);
  }
  __syncthreads();
  // Layer 2: H(128x128) @ W2(128x128) + b2 -> OUT
  {
    v8f acc[8];
#pragma unroll
    for (int r = 0; r < 8; ++r) acc[r] = z;
    gemm_rows<8>(H, LH, W2, HID, 0, w * 16, HID, lane, acc);
#pragma unroll
    for (int r = 0; r < 8; ++r)
      stC(OUT + (size_t)row0 * HID, HID, r * 16, w * 16, lane, acc[r], B2, 0.0f, false);
  }
}

// ---------------------------------------------------------------------------
// Kernel 2: US prefix state for every step (replicates reference recurrence).
// ---------------------------------------------------------------------------
__global__ void us_prefix(const float* __restrict__ us, float* __restrict__ USall) {
  int t = blockIdx.x * blockDim.x + threadIdx.x;
  if (t >= BB * HID) return;
  int b = t >> 7, c = t & 127;
  const float* u = us + (size_t)b * NN * HID + c;
  float s0 = 0.f;
  for (int j = 2; j < NN; ++j) s0 += u[(size_t)j * HID];
  float* o = USall + (size_t)b * NN * HID + c;
  o[0] = s0;
  float run = s0;
  for (int i = 1; i < NN; ++i) {
    run -= u[(size_t)i * HID];
    o[(size_t)i * HID] = (i == NN - 1) ? 0.0f : run;
  }
}

// ---------------------------------------------------------------------------
// Kernel 3: Kcum = cummax(y)+1
// ---------------------------------------------------------------------------
__global__ void prep(const int* __restrict__ y, int* __restrict__ Kcum) {
  if (threadIdx.x == 0 && blockIdx.x == 0) {
    int m = -2147483647;
    for (int i = 0; i < NN; ++i) {
      m = (y[i] > m) ? y[i] : m;
      Kcum[i] = m + 1;
    }
  }
}

// ---------------------------------------------------------------------------
// Kernel 4: one block per (step i, batch b). 8 waves, all GEMMs via f32 WMMA.
// ---------------------------------------------------------------------------
#define SWC_LD 132
#define SA_LD  132
#define SH_LD  516
#define LP_LD  644

__global__ __launch_bounds__(256) void step_kernel(
    const float* __restrict__ wc, const float* __restrict__ USall,
    const int* __restrict__ y, const int* __restrict__ Kcum,
    const float* __restrict__ bcW1, const float* __restrict__ bcB1,
    const float* __restrict__ bcAl, const float* __restrict__ bcW2,
    const float* __restrict__ bcB2, const float* __restrict__ lpW1,
    const float* __restrict__ lpB1, const float* __restrict__ lpA0,
    const float* __restrict__ lpW2, const float* __restrict__ lpB2,
    const float* __restrict__ lpA1, const float* __restrict__ lpW3,
    float* __restrict__ partial) {
  extern __shared__ float sm[];
  int*   sY  = (int*)sm;               // 512 ints  (dynamic-LDS base: offset 0)
  float* sWC = sm + 512;               // 16 x 132
  float* sA  = sWC + 16 * SWC_LD;      // 48 x 132   (later: LP hidden1 32x132)
  float* sH  = sA + 48 * SA_LD;        // 48 x 516   (later: LPin 32x644)
  float* sBC = sH + 48 * SH_LD;        // 48 x 516   (later: LP hidden2 32x132)
  float* sLG = sBC + 48 * SH_LD;       // 32 logits
  float* LPin = sH;
  float* sLH1 = sA;
  float* sLH2 = sBC;

  const int i = blockIdx.x + 1;        // step 1..511
  const int b = blockIdx.y;
  const int t = threadIdx.x;
  const int lane = t & 31, w = t >> 5;
  const float* wcb = wc + (size_t)b * NN * HID;
  const v8f z = {0.f, 0.f, 0.f, 0.f, 0.f, 0.f, 0.f, 0.f};

  // --- stage y via CDNA5 async global->LDS copy (ASYNCcnt-tracked DMA).
  // sY lives at dynamic-LDS offset 0, so the LDS dest address is just j*4.
  for (int j = t; j < NN; j += 256) {
    unsigned lds_addr = (unsigned)(j * 4);
    const int* gaddr = y + j;
    asm volatile("global_load_async_to_lds_b32 %0, %1, off"
                 :: "v"(lds_addr), "v"(gaddr)
                 : "memory");
  }
  asm volatile("s_wait_asynccnt 0x0" ::: "memory");
  __syncthreads();

  // --- WC_i: label-indexed prefix of wc (WC0 has slot0 = wc[:,0]) ---
  {
    int k = t >> 4, cg = t & 15;       // cluster k, 8 channels per thread
    float acc[8];
#pragma unroll
    for (int q = 0; q < 8; ++q) acc[q] = (k == 0) ? wcb[cg * 8 + q] : 0.0f;
    for (int j = 0; j < i; ++j) {
      if (sY[j] == k) {
        const float* p = wcb + (size_t)j * HID + cg * 8;
#pragma unroll
        for (int q = 0; q < 8; ++q) acc[q] += p[q];
      }
    }
#pragma unroll
    for (int q = 0; q < 8; ++q) sWC[k * SWC_LD + cg * 8 + q] = acc[q];
  }
  __syncthreads();

  // --- A (48x128): [WC | WC + wc_cur | wc_cur | zero pad] ---
  const float* wcur = wcb + (size_t)i * HID;
  for (int n = t; n < 48 * HID; n += 256) {
    int r = n >> 7, c = n & 127;
    float v;
    if (r < 16)      v = sWC[r * SWC_LD + c];
    else if (r < 32) v = sWC[(r - 16) * SWC_LD + c] + wcur[c];
    else if (r == 32) v = wcur[c];
    else             v = 0.0f;
    sA[r * SA_LD + c] = v;
  }
  __syncthreads();

  const float aBC = bcAl[0], aL0 = lpA0[0], aL1 = lpA1[0];

  // --- BC layer 1: (48x128) @ (128x512), lrelu -> sH.
  // Wave handles ct strips; 3 row tiles share each B fragment.
  for (int ct = w; ct < 32; ct += 8) {
    v8f acc[3] = {z, z, z};
    gemm_rows<3>(sA, SA_LD, bcW1, BCH, 0, ct * 16, HID, lane, acc);
#pragma unroll
    for (int r = 0; r < 3; ++r)
      stC(sH, SH_LD, r * 16, ct * 16, lane, acc[r], bcB1, aBC, true);
  }
  __syncthreads();
  // --- BC layer 2: (48x512) @ (512x512) -> sBC ---
  for (int ct = w; ct < 32; ct += 8) {
    v8f acc[3] = {z, z, z};
    gemm_rows<3>(sH, SH_LD, bcW2, BCH, 0, ct * 16, BCH, lane, acc);
#pragma unroll
    for (int r = 0; r < 3; ++r)
      stC(sBC, SH_LD, r * 16, ct * 16, lane, acc[r], bcB2, 0.0f, false);
  }
  __syncthreads();

  const int K = Kcum[i - 1];
  const float* USi = USall + ((size_t)b * NN + i) * HID;

  // --- Assemble LPin (17 x 640): BC_fin slots || US broadcast ---
  for (int c = t; c < BCH; c += 256) {
    float S = 0.0f;
    for (int k = 0; k < K; ++k) S += sBC[k * SH_LD + c];
    float bcnew = S + sBC[32 * SH_LD + c];
    for (int s = 0; s <= 16; ++s) {
      float v = (s == 16 || s == K)
                    ? bcnew
                    : (S - sBC[s * SH_LD + c] + sBC[(16 + s) * SH_LD + c]);
      LPin[s * LP_LD + c] = v;
    }
  }
  for (int n = t; n < 32 * HID; n += 256) {
    int s2 = n >> 7, c = n & 127;
    LPin[s2 * LP_LD + BCH + c] = (s2 <= 16) ? USi[c] : 0.0f;
  }
  for (int n = t; n < 15 * BCH; n += 256) {
    int s2 = 17 + n / BCH, c = n % BCH;
    LPin[s2 * LP_LD + c] = 0.0f;
  }
  __syncthreads();

  // --- LP layer 1: (32x640) @ (640x128), lrelu -> sLH1 (ct = wave) ---
  {
    v8f acc[2] = {z, z};
    gemm_rows<2>(LPin, LP_LD, lpW1, HID, 0, w * 16, LPI, lane, acc);
#pragma unroll
    for (int r = 0; r < 2; ++r)
      stC(sLH1, SA_LD, r * 16, w * 16, lane, acc[r], lpB1, aL0, true);
  }
  __syncthreads();
  // --- LP layer 2: (32x128) @ (128x128), lrelu -> sLH2 ---
  {
    v8f acc[2] = {z, z};
    gemm_rows<2>(sLH1, SA_LD, lpW2, HID, 0, w * 16, HID, lane, acc);
#pragma unroll
    for (int r = 0; r < 2; ++r)
      stC(sLH2, SA_LD, r * 16, w * 16, lane, acc[r], lpB2, aL1, true);
  }
  __syncthreads();

  // --- LP layer 3 (128 -> 1, no bias): logits ---
  if (t < 17) {
    float s = 0.0f;
    for (int c = 0; c < HID; ++c) s += sLH2[t * SA_LD + c] * lpW3[c];
    sLG[t] = s;
  }
  __syncthreads();

  // --- masked log-softmax over slots 0..K; partial NLL term ---
  if (t == 0) {
    int ycur = sY[i];
    float mx = -1e30f;
    for (int s = 0; s <= K; ++s) mx = fmaxf(mx, sLG[s]);
    float se = 0.0f;
    for (int s = 0; s <= K; ++s) se += __expf(sLG[s] - mx);
    float logp = sLG[ycur] - mx - __logf(se);
    partial[(size_t)(i - 1) * BB + b] = -logp * (1.0f / (float)BB);
  }
}

// ---------------------------------------------------------------------------
// Kernel 5: deterministic fixed-order reduction of per-step partials.
// ---------------------------------------------------------------------------
__global__ void reduce_nll(const float* __restrict__ partial, float* __restrict__ out) {
  __shared__ float red[256];
  float s = 0.0f;
  for (int n = threadIdx.x; n < (NN - 1) * BB; n += 256) s += partial[n];
  red[threadIdx.x] = s;
  __syncthreads();
  for (int st = 128; st > 0; st >>= 1) {
    if (threadIdx.x < st) red[threadIdx.x] += red[threadIdx.x + st];
    __syncthreads();
  }
  if (threadIdx.x == 0) out[0] = red[0];
}

// ---------------------------------------------------------------------------
// Host launcher
// ---------------------------------------------------------------------------
struct P5 { const float *W1, *W2, *A, *B1, *B2; };

static P5 parse5(void* const* d_in, const int* sz, int base) {
  // pytree order is either sorted keys (Ws, a, bs) or insertion (Ws, bs, a)
  P5 p;
  p.W1 = (const float*)d_in[base];
  p.W2 = (const float*)d_in[base + 1];
  if (sz[base + 2] == 1) {
    p.A = (const float*)d_in[base + 2];
    p.B1 = (const float*)d_in[base + 3];
    p.B2 = (const float*)d_in[base + 4];
  } else {
    p.B1 = (const float*)d_in[base + 2];
    p.B2 = (const float*)d_in[base + 3];
    p.A = (const float*)d_in[base + 4];
  }
  return p;
}

extern "C" void kernel_launch(void* const* d_in, const int* in_sizes, int n_in,
                              void* d_out, int out_size, void* d_ws, size_t ws_size,
                              hipStream_t stream) {
  const float* X = (const float*)d_in[0];
  const int* y = (const int*)d_in[1];
  P5 wcp = parse5(d_in, in_sizes, 2);
  P5 usp = parse5(d_in, in_sizes, 7);
  P5 bcp = parse5(d_in, in_sizes, 12);
  // lp_params: 7 leaves: Ws[0](81920) Ws[1](16384) Ws[2](128), then a/bs in
  // either sorted (a0,a1,b1,b2) or insertion (b1,b2,a0,a1) order.
  const float *lpW1, *lpW2, *lpW3, *lpA0, *lpA1, *lpB1, *lpB2;
  {
    int base = 17;
    lpW1 = (const float*)d_in[base];
    lpW2 = (const float*)d_in[base + 1];
    lpW3 = (const float*)d_in[base + 2];
    if (in_sizes[base + 3] == 1) {
      lpA0 = (const float*)d_in[base + 3];
      lpA1 = (const float*)d_in[base + 4];
      lpB1 = (const float*)d_in[base + 5];
      lpB2 = (const float*)d_in[base + 6];
    } else {
      lpB1 = (const float*)d_in[base + 3];
      lpB2 = (const float*)d_in[base + 4];
      lpA0 = (const float*)d_in[base + 5];
      lpA1 = (const float*)d_in[base + 6];
    }
  }

  char* ws = (char*)d_ws;
  float* wcO   = (float*)(ws);                         // 16*512*128 f32 = 4 MB
  float* usO   = (float*)(ws + ((size_t)4 << 20));     // 4 MB
  float* USall = (float*)(ws + ((size_t)8 << 20));     // 4 MB
  float* part  = (float*)(ws + ((size_t)12 << 20));    // 8176 f32
  int*   Kc    = (int*)(ws + ((size_t)12 << 20) + (64 << 10));

  const size_t featSh = (size_t)128 * 132 * sizeof(float);
  feat_mlp<<<64, 256, featSh, stream>>>(X, wcp.W1, wcp.B1, wcp.A, wcp.W2, wcp.B2, wcO);
  feat_mlp<<<64, 256, featSh, stream>>>(X, usp.W1, usp.B1, usp.A, usp.W2, usp.B2, usO);
  us_prefix<<<8, 256, 0, stream>>>(usO, USall);
  prep<<<1, 64, 0, stream>>>(y, Kc);

  const size_t stepSh =
      (size_t)(512 + 16 * SWC_LD + 48 * SA_LD + 48 * SH_LD + 48 * SH_LD + 32) *
      sizeof(float);
  dim3 grid(NN - 1, BB);
  step_kernel<<<grid, 256, stepSh, stream>>>(wcO, USall, y, Kc,
                                             bcp.W1, bcp.B1, bcp.A, bcp.W2, bcp.B2,
                                             lpW1, lpB1, lpA0, lpW2, lpB2, lpA1, lpW3,
                                             part);
  reduce_nll<<<1, 256, 0, stream>>>(part, (float*)d_out);
}